// DiffusionStar_41841571398360
// MI455X (gfx1250) — compile-verified
//
#include <hip/hip_runtime.h>
#include <math.h>

#define D_FEAT   3072
#define N_TRAIN  100000
#define N_B      32
#define L_STRIDE 100096   // padded row stride for logits (391*256)

typedef __attribute__((ext_vector_type(2))) float v2f;
typedef __attribute__((ext_vector_type(8))) float v8f;

static __device__ __forceinline__ v8f wmma_f32(v2f a, v2f b, v8f c) {
    // D = A(16x4,f32) * B(4x16,f32) + C(16x16,f32)
    return __builtin_amdgcn_wmma_f32_16x16x4_f32(false, a, false, b, (short)0, c,
                                                 false, false);
}

// ---------------------------------------------------------------------------
// K1: logits[b][n] = (s/(1-ab)) * <x_b, t_n>  -  (ab/(2(1-ab))) * ||t_n||^2
//     (row-constant ||x_b||^2 term dropped: cancels in softmax)
// grid: 391 blocks * 256 thr; each wave: 2 column-tiles of 16 train rows,
// both 16-row M tiles of b. One b64 load per fragment per K=4 step.
// ---------------------------------------------------------------------------
__global__ __launch_bounds__(256)
void k1_logits(const float* __restrict__ x, const float* __restrict__ tr,
               const float* __restrict__ acp, const int* __restrict__ tidx,
               float* __restrict__ logits)
{
    const float ab = acp[tidx[0]];
    const float s  = sqrtf(ab);
    const float om = 1.0f - ab;
    const float cs = s / om;
    const float ts = ab / (2.0f * om);

    const int lane = threadIdx.x & 31;
    const int wave = threadIdx.x >> 5;
    const int ln16 = lane & 15;
    const int kh   = lane >> 4;          // 0: K{0,1}, 1: K{2,3}
    const int koff = kh * 2;

    for (int i = 0; i < 2; ++i) {
        const int ct = blockIdx.x * 16 + wave * 2 + i;   // 16-column tile index
        if (ct >= N_TRAIN / 16) continue;                // wave-uniform
        const int nbase = ct * 16;
        const int ncol  = nbase + ln16;

        const float* xrow0 = x  + (size_t)ln16        * D_FEAT;
        const float* xrow1 = x  + (size_t)(ln16 + 16) * D_FEAT;
        const float* trow  = tr + (size_t)ncol        * D_FEAT;

        v8f c0 = {}; v8f c1 = {};
        float tsq = 0.0f;
        #pragma unroll 4
        for (int k = 0; k < D_FEAT; k += 4) {
            v2f a0  = *(const v2f*)(xrow0 + k + koff);
            v2f a1  = *(const v2f*)(xrow1 + k + koff);
            v2f bfr = *(const v2f*)(trow  + k + koff);
            tsq = fmaf(bfr.x, bfr.x, tsq);
            tsq = fmaf(bfr.y, bfr.y, tsq);
            c0 = wmma_f32(a0, bfr, c0);
            c1 = wmma_f32(a1, bfr, c1);
        }
        tsq += __shfl_xor(tsq, 16, 32);  // combine the two K-halves per column

        #pragma unroll
        for (int j = 0; j < 8; ++j) {    // D layout: VGPR j -> M = j + 8*(lane>>4)
            const int m = j + kh * 8;
            logits[(size_t)m        * L_STRIDE + ncol] = cs * c0[j] - ts * tsq;
            logits[(size_t)(m + 16) * L_STRIDE + ncol] = cs * c1[j] - ts * tsq;
        }
    }
}

// ---------------------------------------------------------------------------
// K2: per-row softmax stats: ml[b] = rowmax, ml[32+b] = 1/sum(exp(.-max))
// ---------------------------------------------------------------------------
__global__ __launch_bounds__(256)
void k2_stats(const float* __restrict__ logits, float* __restrict__ ml)
{
    __shared__ float red[256];
    const int b = blockIdx.x;
    const float* row = logits + (size_t)b * L_STRIDE;

    float mx = -3.4e38f;
    for (int n = threadIdx.x; n < N_TRAIN; n += 256) mx = fmaxf(mx, row[n]);
    red[threadIdx.x] = mx; __syncthreads();
    for (int st = 128; st > 0; st >>= 1) {
        if ((int)threadIdx.x < st)
            red[threadIdx.x] = fmaxf(red[threadIdx.x], red[threadIdx.x + st]);
        __syncthreads();
    }
    mx = red[0]; __syncthreads();

    float sm = 0.0f;
    for (int n = threadIdx.x; n < N_TRAIN; n += 256) sm += __expf(row[n] - mx);
    red[threadIdx.x] = sm; __syncthreads();
    for (int st = 128; st > 0; st >>= 1) {
        if ((int)threadIdx.x < st) red[threadIdx.x] += red[threadIdx.x + st];
        __syncthreads();
    }
    if (threadIdx.x == 0) { ml[b] = mx; ml[32 + b] = 1.0f / red[0]; }
}

// ---------------------------------------------------------------------------
// K3: out[b][d] = x[b][d]*inv - s*inv * sum_n w[b][n]*tr[n][d],
//     w = exp(logit - m_b) / l_b.  Block owns 16 feature columns; 8 waves
//     split N (12500 each), LDS-reduce partial C tiles, fused epilogue.
// ---------------------------------------------------------------------------
__global__ __launch_bounds__(256)
void k3_weighted(const float* __restrict__ x, const float* __restrict__ tr,
                 const float* __restrict__ acp, const int* __restrict__ tidx,
                 const float* __restrict__ logits, const float* __restrict__ ml,
                 float* __restrict__ out)
{
    const float ab   = acp[tidx[0]];
    const float s    = sqrtf(ab);
    const float om   = 1.0f - ab;
    const float inv  = rsqrtf(om);
    const float sinv = s * inv;

    const int lane = threadIdx.x & 31;
    const int wave = threadIdx.x >> 5;
    const int ln16 = lane & 15;
    const int kh   = lane >> 4;

    const int d0   = blockIdx.x * 16;
    const int dcol = d0 + ln16;

    const float m0v = ml[ln16];
    const float il0 = ml[32 + ln16];
    const float m1v = ml[ln16 + 16];
    const float il1 = ml[32 + ln16 + 16];

    const int chunk = N_TRAIN / 8;      // 12500, multiple of 4
    const int n0    = wave * chunk;

    const float* lrow0 = logits + (size_t)ln16        * L_STRIDE;
    const float* lrow1 = logits + (size_t)(ln16 + 16) * L_STRIDE;

    v8f c0 = {}; v8f c1 = {};
    for (int nb = 0; nb < chunk; nb += 4) {
        const int kn = n0 + nb + kh * 2;
        v2f l0 = *(const v2f*)(lrow0 + kn);
        v2f l1 = *(const v2f*)(lrow1 + kn);
        v2f a0, a1, bfr;
        a0.x = __expf(l0.x - m0v) * il0;
        a0.y = __expf(l0.y - m0v) * il0;
        a1.x = __expf(l1.x - m1v) * il1;
        a1.y = __expf(l1.y - m1v) * il1;
        // B[k][d] = tr[kn+vgpr][d]; stream train nontemporally (single use)
        bfr.x = __builtin_nontemporal_load(tr + (size_t)kn       * D_FEAT + dcol);
        bfr.y = __builtin_nontemporal_load(tr + (size_t)(kn + 1) * D_FEAT + dcol);
        c0 = wmma_f32(a0, bfr, c0);
        c1 = wmma_f32(a1, bfr, c1);
    }

    __shared__ float buf[8][512];
    #pragma unroll
    for (int j = 0; j < 8; ++j) {
        buf[wave][      j * 32 + lane] = c0[j];
        buf[wave][256 + j * 32 + lane] = c1[j];
    }
    __syncthreads();

    for (int o = threadIdx.x; o < 512; o += 256) {
        float v = 0.0f;
        #pragma unroll
        for (int w = 0; w < 8; ++w) v += buf[w][o];
        const int tile = o >> 8;
        const int j    = (o >> 5) & 7;
        const int ln   = o & 31;
        const int m    = j + ((ln >> 4) << 3) + tile * 16;
        const int d    = d0 + (ln & 15);
        out[(size_t)m * D_FEAT + d] = x[(size_t)m * D_FEAT + d] * inv - sinv * v;
    }
}

extern "C" void kernel_launch(void* const* d_in, const int* in_sizes, int n_in,
                              void* d_out, int out_size, void* d_ws, size_t ws_size,
                              hipStream_t stream) {
    (void)in_sizes; (void)n_in; (void)out_size; (void)ws_size;
    const float* x   = (const float*)d_in[0];
    const float* tr  = (const float*)d_in[1];
    const float* acp = (const float*)d_in[2];
    const int*   t   = (const int*)d_in[3];
    float* out    = (float*)d_out;
    float* logits = (float*)d_ws;                         // 32 * 100096 f32
    float* ml     = logits + (size_t)N_B * L_STRIDE;      // 64 f32

    k1_logits  <<<391, 256, 0, stream>>>(x, tr, acp, t, logits);
    k2_stats   <<< 32, 256, 0, stream>>>(logits, ml);
    k3_weighted<<<192, 256, 0, stream>>>(x, tr, acp, t, logits, ml, out);
}